// RNNcell_57312043598298
// MI455X (gfx1250) — compile-verified
//
#include <hip/hip_runtime.h>

typedef __attribute__((ext_vector_type(2))) float v2f;
typedef __attribute__((ext_vector_type(8))) float v8f;

#define SEQ_LEN 512
#define BATCH   64
#define IN_DIM  512
#define HID     512

// WMMA f32 16x16x4 fragment layout (CDNA5 ISA §7.12.2):
//   A/B element (x, k): lane = x + 16*((k>>1)&1), vgpr = k & 1   (k in 0..3)
//   C/D element (m, n): lane = n + 16*(m>=8),     vgpr = m & 7
static __device__ __forceinline__ v8f wmma4(v2f a, v2f b, v8f c) {
    return __builtin_amdgcn_wmma_f32_16x16x4_f32(
        false, a, false, b, (short)0, c, false, false);
}

// ---------------------------------------------------------------------------
// Kernel 1: xw = input @ W_ih + bias.  M=32768, K=512, N=512.
// 2x2 register blocking: one wave computes a 32x32 output block (4 tiles),
// reusing each A/B fragment twice -> 1 fragment load per WMMA instead of 2.
// ---------------------------------------------------------------------------
__global__ __launch_bounds__(256) void xw_gemm_kernel(
    const float* __restrict__ x,      // [M, K] row-major
    const float* __restrict__ w,      // [K, N] row-major
    const float* __restrict__ bias,   // [N]
    float* __restrict__ xw)           // [M, N] row-major
{
    const int M  = SEQ_LEN * BATCH;
    const int NB = HID / 32;                                  // 16 N-blocks
    int wave = (int)((blockIdx.x * blockDim.x + threadIdx.x) >> 5);
    int lane = threadIdx.x & 31;
    int xl   = lane & 15;
    int hi   = lane >> 4;

    int m0 = (wave / NB) * 32;
    int n0 = (wave % NB) * 32;
    if (m0 >= M) return;

    const float* arow0 = x + (size_t)(m0 + xl) * IN_DIM + 2 * hi;
    const float* arow1 = arow0 + (size_t)16 * IN_DIM;
    const float* bcol0 = w + (size_t)(2 * hi) * HID + n0 + xl;
    const float* bcol1 = bcol0 + 16;

    v8f c00 = {}, c01 = {}, c10 = {}, c11 = {};
    for (int kk = 0; kk < IN_DIM; kk += 4) {
        v2f a0; a0.x = arow0[kk];               a0.y = arow0[kk + 1];
        v2f a1; a1.x = arow1[kk];               a1.y = arow1[kk + 1];
        v2f b0; b0.x = bcol0[(size_t)kk * HID]; b0.y = bcol0[(size_t)(kk + 1) * HID];
        v2f b1; b1.x = bcol1[(size_t)kk * HID]; b1.y = bcol1[(size_t)(kk + 1) * HID];
        c00 = wmma4(a0, b0, c00);
        c01 = wmma4(a0, b1, c01);
        c10 = wmma4(a1, b0, c10);
        c11 = wmma4(a1, b1, c11);
    }

    float bv0 = bias[n0 + xl];
    float bv1 = bias[n0 + 16 + xl];
#pragma unroll
    for (int j = 0; j < 8; ++j) {
        size_t r0 = (size_t)(m0 + j + 8 * hi) * HID;
        size_t r1 = (size_t)(m0 + 16 + j + 8 * hi) * HID;
        xw[r0 + n0 + xl]      = c00[j] + bv0;
        xw[r0 + n0 + 16 + xl] = c01[j] + bv1;
        xw[r1 + n0 + xl]      = c10[j] + bv0;
        xw[r1 + n0 + 16 + xl] = c11[j] + bv1;
    }
}

// ---------------------------------------------------------------------------
// Kernel 1b (one-time, 1 MB): repack W_hh into WMMA-fragment-major order.
// Flat index f = (((n*64 + q2)*32) + l)*4 + v2 holds
//   whh[k*HID + n*16 + (l&15)]  with  k = 8*q2 + 4*(v2>>1) + 2*(l>>4) + (v2&1)
// so in the scan each lane reads ONE float4 (global_load_b128, 512 B fully
// coalesced per wave) per two WMMA K-steps instead of four strided b32 loads.
// ---------------------------------------------------------------------------
__global__ __launch_bounds__(256) void pack_whh_kernel(
    const float* __restrict__ whh,    // [H, H] row-major
    float* __restrict__ wp)           // [H*H] packed
{
    int f = (int)(blockIdx.x * blockDim.x + threadIdx.x);
    if (f >= HID * HID) return;
    int v2 = f & 3;
    int l  = (f >> 2) & 31;
    int q2 = (f >> 7) & 63;
    int n  = f >> 13;
    int k  = 8 * q2 + 4 * (v2 >> 1) + 2 * (l >> 4) + (v2 & 1);
    int col = n * 16 + (l & 15);
    wp[f] = whh[(size_t)k * HID + col];
}

// ---------------------------------------------------------------------------
// Kernel 2: sequential scan  h = tanh(xw_t + h @ W_hh)
// Batch rows are independent -> 4 workgroups x 16 batch rows, zero inter-WG
// communication. h chunk is DOUBLE-BUFFERED in LDS (read buf[t&1], write the
// other) so only ONE workgroup barrier per step is on the serial chain.
// Padded stride 516 keeps the ds_load_b64 A-fragments bank-conflict-free.
// xw_t loads are issued BEFORE the WMMA K-loop so their latency hides under
// the 128-WMMA accumulation chain. Packed W_hh streams from L2 as 64
// coalesced b128 loads per wave per step.
// ---------------------------------------------------------------------------
__global__ __launch_bounds__(1024) void rnn_scan_kernel(
    const float* __restrict__ xw,     // [T, B, H] (+bias) or nullptr
    const float* __restrict__ h0,     // [B, H]
    const float* __restrict__ whh,    // [H, H] row-major (strided fallback)
    const float* __restrict__ wpack,  // packed W_hh or nullptr
    const float* __restrict__ xin,    // [T, B, IN] (fallback)
    const float* __restrict__ wih,    // [IN, H]    (fallback)
    const float* __restrict__ bias,   // [H]        (fallback)
    float* __restrict__ out)          // [2, B, H]
{
    constexpr int HS = HID + 4;       // 516: conflict-free padded LDS stride
    __shared__ float hbuf[2][16][HS]; // ~66 KB (gfx1250: 320 KB LDS per WGP)

    const int g    = blockIdx.x;      // batch chunk: rows 16g .. 16g+15
    const int tid  = threadIdx.x;
    const int wave = tid >> 5;        // 0..31 -> N tile
    const int lane = tid & 31;
    const int xl   = lane & 15;
    const int hi   = lane >> 4;
    const int n0   = wave * 16;

    for (int i = tid; i < 16 * HID; i += (int)blockDim.x) {
        int r = i >> 9, col = i & 511;
        hbuf[0][r][col] = h0[(size_t)(g * 16 + r) * HID + col];
    }
    __syncthreads();

    const float*  bcol  = whh + (size_t)(2 * hi) * HID + n0 + xl;
    const float4* bp    = wpack ? (const float4*)wpack + ((size_t)wave * 64) * 32 + lane
                                : nullptr;
    const float   biasv = (xw == nullptr) ? bias[n0 + xl] : 0.0f;

    for (int t = 0; t < SEQ_LEN; ++t) {
        const int rd = t & 1;
        const int wr = 1 - rd;

        // Issue the xw_t loads first: independent of the K-loop, their L2
        // latency is hidden under the WMMA accumulation chain.
        float base[8];
        if (xw) {
            const float* xwt = xw + ((size_t)t * BATCH + (size_t)g * 16) * HID;
#pragma unroll
            for (int j = 0; j < 8; ++j)
                base[j] = xwt[(size_t)(j + 8 * hi) * HID + n0 + xl];
        } else {
#pragma unroll
            for (int j = 0; j < 8; ++j) base[j] = biasv;
        }

        v8f c = {};

        if (xw == nullptr) {
            // Fallback: accumulate x_t @ W_ih into c first (strided B loads).
            const float* axr = xin + ((size_t)t * BATCH + g * 16 + xl) * IN_DIM + 2 * hi;
            const float* bxc = wih + (size_t)(2 * hi) * HID + n0 + xl;
            for (int kk = 0; kk < IN_DIM; kk += 4) {
                v2f a; a.x = axr[kk]; a.y = axr[kk + 1];
                v2f b; b.x = bxc[(size_t)kk * HID]; b.y = bxc[(size_t)(kk + 1) * HID];
                c = wmma4(a, b, c);
            }
        }

        // c += h @ W_hh
        if (bp) {
            // Packed path: one coalesced b128 per two WMMA K-steps.
            for (int q2 = 0; q2 < 64; ++q2) {
                float4 bq = bp[(size_t)q2 * 32];
                int kk = 8 * q2;
                v2f a0; a0.x = hbuf[rd][xl][kk + 2 * hi];
                        a0.y = hbuf[rd][xl][kk + 2 * hi + 1];
                v2f a1; a1.x = hbuf[rd][xl][kk + 4 + 2 * hi];
                        a1.y = hbuf[rd][xl][kk + 4 + 2 * hi + 1];
                v2f b0; b0.x = bq.x; b0.y = bq.y;
                v2f b1; b1.x = bq.z; b1.y = bq.w;
                c = wmma4(a0, b0, c);
                c = wmma4(a1, b1, c);
            }
        } else {
            for (int kk = 0; kk < HID; kk += 4) {
                v2f a; a.x = hbuf[rd][xl][kk + 2 * hi];
                       a.y = hbuf[rd][xl][kk + 2 * hi + 1];
                v2f b; b.x = bcol[(size_t)kk * HID];
                       b.y = bcol[(size_t)(kk + 1) * HID];
                c = wmma4(a, b, c);
            }
        }

        // h_new = tanh(xw_t + c) -> write into the OTHER buffer (no WAR
        // hazard), so a single barrier per step suffices.
#pragma unroll
        for (int j = 0; j < 8; ++j)
            hbuf[wr][j + 8 * hi][n0 + xl] = tanhf(c[j] + base[j]);

        __syncthreads();
    }

    // SEQ_LEN is even -> final state is in buffer 0.
    for (int i = tid; i < 16 * HID; i += (int)blockDim.x) {
        int r = i >> 9, col = i & 511;
        float v = hbuf[0][r][col];
        size_t o = (size_t)(g * 16 + r) * HID + col;
        out[o] = v;
        out[o + (size_t)BATCH * HID] = v;
    }
}

// ---------------------------------------------------------------------------
extern "C" void kernel_launch(void* const* d_in, const int* in_sizes, int n_in,
                              void* d_out, int out_size, void* d_ws, size_t ws_size,
                              hipStream_t stream) {
    const float* x    = (const float*)d_in[0];   // [T, B, IN]
    const float* h0   = (const float*)d_in[1];   // [B, H]
    const float* wih  = (const float*)d_in[2];   // [IN, H]
    const float* whh  = (const float*)d_in[3];   // [H, H]
    const float* bias = (const float*)d_in[4];   // [H]
    float* out = (float*)d_out;

    const size_t xw_bytes = (size_t)SEQ_LEN * BATCH * HID * sizeof(float); // 64 MB
    const size_t wp_bytes = (size_t)HID * HID * sizeof(float);             // 1 MB
    char* ws = (char*)d_ws;

    float* xw = nullptr;
    float* wp = nullptr;
    if (ws_size >= xw_bytes + wp_bytes) {
        xw = (float*)ws;
        wp = (float*)(ws + xw_bytes);
    } else if (ws_size >= xw_bytes) {
        xw = (float*)ws;
    } else if (ws_size >= wp_bytes) {
        wp = (float*)ws;
    }

    if (xw) {
        const int n_waves = (SEQ_LEN * BATCH / 32) * (HID / 32);   // 16384
        xw_gemm_kernel<<<n_waves / 8, 256, 0, stream>>>(x, wih, bias, xw);
    }
    if (wp) {
        pack_whh_kernel<<<(HID * HID) / 256, 256, 0, stream>>>(whh, wp);
    }
    rnn_scan_kernel<<<BATCH / 16, 1024, 0, stream>>>(
        xw, h0, whh, wp, x, wih, bias, out);
}